// Mamba_60043642798522
// MI455X (gfx1250) — compile-verified
//
#include <hip/hip_runtime.h>
#include <hip/hip_bf16.h>

// ---------------------------------------------------------------------------
// Mamba forward for MI455X (gfx1250, wave32).
//  - Big GEMMs: 128x128 block tiles staged through LDS (single buffer,
//    register prefetch of next K-chunk), 8 waves/block, 32x64 wave tiles,
//    V_WMMA_F32_16X16X4_F32 (full fp32 accuracy vs fp32 reference).
//    __launch_bounds__(256,1) gives the allocator the full VGPR budget —
//    round-2 codegen spilled the prefetch registers to scratch inside the
//    hot loop (scratch_load_b128 reloads); 2 waves/SIMD is enough occupancy
//    since latency is hidden by the dscnt-pipelined WMMA stream.
//  - Small per-head GEMMs: direct register WMMA.
//  - Selective scan: one thread per (b,h,d,n) state element, 16-lane
//    __shfl_xor reduction for y, gating fused into the epilogue.
// ---------------------------------------------------------------------------

typedef __attribute__((ext_vector_type(2))) float v2f;
typedef __attribute__((ext_vector_type(8))) float v8f;

#define D_MODEL  1024
#define N_HEADS  4
#define D_CONV   4
#define D_INNER  2048
#define H_DIN    512
#define H_DT     16
#define H_DS     16
#define BATCH    2
#define SEQ      1024
#define NTOK     (BATCH * SEQ)      // 2048 tokens
#define XO       (2 * H_DS + H_DT)  // 48 outputs per head in x_dbl
#define LDS_PAD  20                 // row stride (floats): keeps float4 align,
                                    // near-conflict-free frag reads

// ---------------------------------------------------------------------------
// Direct (register-only) wave-level fp32 WMMA tile for the small GEMMs:
// D[16 x 16*NT] += A[16 x K] * W^T, A row-major (lda), W row-major N x K (ldb).
// A-frag: lanes 0-15 rows M=lane, VGPR0/1 = k,k+1; lanes 16-31 = k+2,k+3.
// B-frag mirrors it per ISA B-layout.
// ---------------------------------------------------------------------------
template <int NT>
__device__ __forceinline__ void wmma_tile_f32(const float* __restrict__ A, int lda,
                                              const float* __restrict__ W, int ldb,
                                              int K, v8f* acc) {
    const int lane = threadIdx.x & 31;
    const int hh   = lane >> 4;
    const int r    = lane & 15;
    const float* ap = A + r * lda + 2 * hh;
    const float* bp = W + r * ldb + 2 * hh;
    for (int k = 0; k < K; k += 4) {
        v2f a = *(const v2f*)(ap + k);
#pragma unroll
        for (int j = 0; j < NT; ++j) {
            v2f b = *(const v2f*)(bp + 16 * j * ldb + k);
            acc[j] = __builtin_amdgcn_wmma_f32_16x16x4_f32(
                false, a, false, b, (short)0, acc[j], false, false);
        }
    }
}

// D layout: VGPR i -> (row i + 8*half, col lane&15)
template <int NT>
__device__ __forceinline__ void store_tile(float* __restrict__ C, int ldc, const v8f* acc) {
    const int lane = threadIdx.x & 31;
    const int hh   = lane >> 4;
    const int r    = lane & 15;
#pragma unroll
    for (int j = 0; j < NT; ++j)
#pragma unroll
        for (int i = 0; i < 8; ++i)
            C[(i + 8 * hh) * ldc + 16 * j + r] = acc[j][i];
}

// ---------------------------------------------------------------------------
// LDS-staged block GEMM:  C[M x N] = A[M x K] @ W[N x K]^T
// Block tile 128x128, BK=16, 256 threads = 8 waves as 4(M) x 2(N),
// wave tile 32x64 (2 x 4 WMMA accumulators).  M is fixed at 2048 (16 blocks).
// ---------------------------------------------------------------------------
__global__ void __launch_bounds__(256, 1)
k_gemm128(const float* __restrict__ A, const float* __restrict__ W,
          float* __restrict__ C, int K, int ldc) {
    __shared__ float ldsA[128 * LDS_PAD];
    __shared__ float ldsB[128 * LDS_PAD];

    const int m_blk = blockIdx.x & 15;    // M/128 == 16
    const int n_blk = blockIdx.x >> 4;
    const int tid   = threadIdx.x;
    const int wave  = tid >> 5;
    const int lane  = tid & 31;
    const int wm    = wave >> 1;          // 0..3 -> m offset wm*32
    const int wn    = wave & 1;           // 0..1 -> n offset wn*64
    const int hh    = lane >> 4;
    const int r     = lane & 15;

    // global->LDS staging: each thread owns half a row (8 floats) of A and W
    const int lrow = tid >> 1;            // 0..127
    const int lcol = (tid & 1) * 8;       // 0 or 8
    const float* ag = A + (m_blk * 128 + lrow) * K + lcol;
    const float* bg = W + (n_blk * 128 + lrow) * K + lcol;

    v8f acc[2][4] = {};

    float4 ra0 = *(const float4*)(ag);
    float4 ra1 = *(const float4*)(ag + 4);
    float4 rb0 = *(const float4*)(bg);
    float4 rb1 = *(const float4*)(bg + 4);

    const int NK = K >> 4;
    for (int kc = 0; kc < NK; ++kc) {
        *(float4*)&ldsA[lrow * LDS_PAD + lcol]     = ra0;
        *(float4*)&ldsA[lrow * LDS_PAD + lcol + 4] = ra1;
        *(float4*)&ldsB[lrow * LDS_PAD + lcol]     = rb0;
        *(float4*)&ldsB[lrow * LDS_PAD + lcol + 4] = rb1;
        __syncthreads();

        if (kc + 1 < NK) {                 // prefetch next chunk into VGPRs,
            const float* agn = ag + (kc + 1) * 16;   // overlaps WMMA below
            const float* bgn = bg + (kc + 1) * 16;
            ra0 = *(const float4*)(agn);
            ra1 = *(const float4*)(agn + 4);
            rb0 = *(const float4*)(bgn);
            rb1 = *(const float4*)(bgn + 4);
        }

#pragma unroll
        for (int kk = 0; kk < 4; ++kk) {
            v2f af[2], bf[4];
#pragma unroll
            for (int i = 0; i < 2; ++i)
                af[i] = *(const v2f*)&ldsA[(wm * 32 + i * 16 + r) * LDS_PAD + kk * 4 + 2 * hh];
#pragma unroll
            for (int j = 0; j < 4; ++j)
                bf[j] = *(const v2f*)&ldsB[(wn * 64 + j * 16 + r) * LDS_PAD + kk * 4 + 2 * hh];
#pragma unroll
            for (int i = 0; i < 2; ++i)
#pragma unroll
                for (int j = 0; j < 4; ++j)
                    acc[i][j] = __builtin_amdgcn_wmma_f32_16x16x4_f32(
                        false, af[i], false, bf[j], (short)0, acc[i][j], false, false);
        }
        __syncthreads();                   // protect LDS before next stage
    }

#pragma unroll
    for (int i = 0; i < 2; ++i)
#pragma unroll
        for (int j = 0; j < 4; ++j)
#pragma unroll
            for (int vi = 0; vi < 8; ++vi) {
                const int row = m_blk * 128 + wm * 32 + i * 16 + vi + 8 * hh;
                const int col = n_blk * 128 + wn * 64 + j * 16 + r;
                C[row * ldc + col] = acc[i][j][vi];
            }
}

// ---------------------------------------------------------------------------
// Stage 2: causal depthwise conv (width 4) + bias + SiLU on the x half of xz.
// ---------------------------------------------------------------------------
__global__ void k_conv_silu(const float* __restrict__ XZ, const float* __restrict__ CW,
                            const float* __restrict__ CB, float* __restrict__ XC) {
    const int idx = blockIdx.x * blockDim.x + threadIdx.x;
    if (idx >= NTOK * D_INNER) return;
    const int c = idx & (D_INNER - 1);
    const int t = idx >> 11;
    const int l = t & (SEQ - 1);
    const int b = t >> 10;
    float acc = CB[c];
#pragma unroll
    for (int k = 0; k < D_CONV; ++k) {
        const int ls = l - (D_CONV - 1) + k;
        if (ls >= 0)
            acc = fmaf(XZ[(b * SEQ + ls) * (2 * D_INNER) + c], CW[c * D_CONV + k], acc);
    }
    XC[idx] = acc / (1.0f + __expf(-acc));   // SiLU
}

// ---------------------------------------------------------------------------
// Stage 3: x_dbl[t, h, 0..48) = xh[t, h, :512] @ xw[h]^T   (16x48 wave tiles)
// ---------------------------------------------------------------------------
__global__ void k_xdbl(const float* __restrict__ XC, const float* __restrict__ XPW,
                       float* __restrict__ XDBL) {
    const int wave   = (blockIdx.x * blockDim.x + threadIdx.x) >> 5;
    const int m_tile = wave & 127;
    const int h      = wave >> 7;
    v8f acc[3] = {};
    wmma_tile_f32<3>(XC + m_tile * 16 * D_INNER + h * H_DIN, D_INNER,
                     XPW + h * XO * H_DIN, H_DIN, H_DIN, acc);
    store_tile<3>(XDBL + m_tile * 16 * (N_HEADS * XO) + h * XO, N_HEADS * XO, acc);
}

// ---------------------------------------------------------------------------
// Stage 4: delta = softplus( dt_r @ dtw[h]^T + dt_bias ),  K=16.
// ---------------------------------------------------------------------------
__global__ void k_delta(const float* __restrict__ XDBL, const float* __restrict__ DTW,
                        const float* __restrict__ DTB, float* __restrict__ DELTA) {
    const int wave   = (blockIdx.x * blockDim.x + threadIdx.x) >> 5;
    const int m_tile = wave & 127;
    const int n_tile = (wave >> 7) & 7;
    const int h      = wave >> 10;
    v8f acc[4] = {};
    wmma_tile_f32<4>(XDBL + m_tile * 16 * (N_HEADS * XO) + h * XO, N_HEADS * XO,
                     DTW + (h * H_DIN + n_tile * 64) * H_DT, H_DT, H_DT, acc);
    const int lane = threadIdx.x & 31;
    const int hh   = lane >> 4;
    const int r    = lane & 15;
#pragma unroll
    for (int j = 0; j < 4; ++j)
#pragma unroll
        for (int i = 0; i < 8; ++i) {
            const int col = h * H_DIN + n_tile * 64 + 16 * j + r;
            const int row = m_tile * 16 + i + 8 * hh;
            float v = acc[j][i] + DTB[col];
            v = (v > 20.0f) ? v : log1pf(__expf(v));   // softplus
            DELTA[row * D_INNER + col] = v;
        }
}

// ---------------------------------------------------------------------------
// Stage 5: selective scan.  One thread per state element (b,h,d,n):
// 65536 threads = 2048 waves (~2/SIMD).  Serial chain per step is
// 1 exp + 2 fma + 4-step shfl_xor reduction over the 16 n-lanes.
// Gating  yg = (y + u*D) * silu(z)  fused (n==0 lanes write).
// Grid: 256 blocks (b*h*dchunk) x 256 threads (16 d x 16 n).
// ---------------------------------------------------------------------------
__global__ void k_scan(const float* __restrict__ XZ, const float* __restrict__ XC,
                       const float* __restrict__ XDBL, const float* __restrict__ DELTA,
                       const float* __restrict__ ALOG, const float* __restrict__ DSK,
                       float* __restrict__ YG) {
    const int blk    = blockIdx.x;
    const int dchunk = blk & 31;          // H_DIN/16
    const int h      = (blk >> 5) & 3;
    const int b      = blk >> 7;
    const int tid    = threadIdx.x;
    const int n      = tid & 15;
    const int d      = dchunk * 16 + (tid >> 4);
    const int c      = h * H_DIN + d;

    const float a   = -__expf(ALOG[c * H_DS + n]);
    const float dsk = DSK[c];
    float s = 0.0f;

    for (int l = 0; l < SEQ; ++l) {
        const int t = b * SEQ + l;
        const float dt = DELTA[t * D_INNER + c];       // broadcast in 16-lane grp
        const float u  = XC[t * D_INNER + c];
        const float bv = XDBL[t * (N_HEADS * XO) + h * XO + H_DT + n];
        const float cv = XDBL[t * (N_HEADS * XO) + h * XO + H_DT + H_DS + n];

        const float dA = __expf(dt * a);
        s = fmaf(s, dA, dt * u * bv);
        float y = s * cv;
        y += __shfl_xor(y, 1);
        y += __shfl_xor(y, 2);
        y += __shfl_xor(y, 4);
        y += __shfl_xor(y, 8);
        if (n == 0) {
            const float z  = XZ[t * (2 * D_INNER) + D_INNER + c];
            const float sz = z / (1.0f + __expf(-z));  // silu(z)
            YG[t * D_INNER + c] = (y + u * dsk) * sz;
        }
    }
}

// ---------------------------------------------------------------------------
extern "C" void kernel_launch(void* const* d_in, const int* in_sizes, int n_in,
                              void* d_out, int out_size, void* d_ws, size_t ws_size,
                              hipStream_t stream) {
    const float* Hs   = (const float*)d_in[0];  // hidden_states (2,1024,1024)
    const float* Wi   = (const float*)d_in[1];  // in_proj_w     (4096,1024)
    const float* CW   = (const float*)d_in[2];  // conv_w        (2048,1,4)
    const float* CB   = (const float*)d_in[3];  // conv_b        (2048)
    const float* XPW  = (const float*)d_in[4];  // x_proj_w      (192,512)
    const float* DTW  = (const float*)d_in[5];  // dt_proj_w     (2048,16)
    const float* DTB  = (const float*)d_in[6];  // dt_proj_b     (2048)
    const float* ALOG = (const float*)d_in[7];  // A_log         (2048,16)
    const float* DSK  = (const float*)d_in[8];  // Dskip         (2048)
    const float* WO   = (const float*)d_in[9];  // out_proj_w    (1024,2048)
    float* out = (float*)d_out;

    float* ws    = (float*)d_ws;
    float* XZ    = ws;                                   // 2048 x 4096
    float* XC    = XZ    + (size_t)NTOK * 2 * D_INNER;   // 2048 x 2048
    float* XDBL  = XC    + (size_t)NTOK * D_INNER;       // 2048 x 192
    float* DELTA = XDBL  + (size_t)NTOK * N_HEADS * XO;  // 2048 x 2048
    float* YG    = DELTA + (size_t)NTOK * D_INNER;       // 2048 x 2048

    // in_proj: M=2048, N=4096, K=1024 -> 16 x 32 = 512 blocks
    k_gemm128<<<512, 256, 0, stream>>>(Hs, Wi, XZ, D_MODEL, 2 * D_INNER);
    k_conv_silu<<<(NTOK * D_INNER + 255) / 256, 256, 0, stream>>>(XZ, CW, CB, XC);
    // 512 waves -> 64 blocks x 256
    k_xdbl<<<64, 256, 0, stream>>>(XC, XPW, XDBL);
    // 4096 waves -> 512 blocks x 256
    k_delta<<<512, 256, 0, stream>>>(XDBL, DTW, DTB, DELTA);
    // scan: 256 blocks x 256 threads
    k_scan<<<256, 256, 0, stream>>>(XZ, XC, XDBL, DELTA, ALOG, DSK, YG);
    // out_proj: M=2048, N=1024, K=2048 -> 16 x 8 = 128 blocks
    k_gemm128<<<128, 256, 0, stream>>>(YG, WO, out, D_INNER, D_MODEL);
}